// MambaHybridLocal_49941879718199
// MI455X (gfx1250) — compile-verified
//
#include <hip/hip_runtime.h>
#include <hip/hip_bf16.h>
#include <math.h>

// ---------------------------------------------------------------------------
// MambaHybridLocal for MI455X (gfx1250), fp32 end-to-end.
// GEMMs: V_WMMA_F32_16X16X4_F32 fed from double-buffered LDS panels staged
// with GLOBAL_LOAD_ASYNC_TO_LDS_B128 (ASYNCcnt) — 8-wave blocks, 16x128 strip,
// async copy of chunk c+1 overlapped with WMMA consumption of chunk c.
// ---------------------------------------------------------------------------

typedef float v2f __attribute__((ext_vector_type(2)));
typedef float v8f __attribute__((ext_vector_type(8)));

#define Lq   512
#define DMq  320
#define DIq  640
#define DSq  128
#define DTRq 20
#define DBLq 276   // DTR + 2*DS
#define NHq  4
#define DHq  80
#define WINq 8

#define BNq  128   // output columns per block (8 waves x 16)
#define KCq  64    // K chunk staged in LDS
#define LDSS 68    // padded row stride (floats): banks 4*lane, conflict-free

enum { OP_NONE = 0, OP_SOFTPLUS = 1 };

// ---------------- wave / block reductions (wave32) -------------------------
__device__ __forceinline__ float warpReduceSum(float v) {
#pragma unroll
    for (int o = 16; o > 0; o >>= 1) v += __shfl_xor(v, o, 32);
    return v;
}

__device__ __forceinline__ float blockReduceSum(float v, float* red) {
    const int lane = threadIdx.x & 31;
    const int wid  = threadIdx.x >> 5;
    float s = warpReduceSum(v);
    __syncthreads();                 // protect 'red' against prior use
    if (lane == 0) red[wid] = s;
    __syncthreads();
    const int nw = blockDim.x >> 5;
    if (wid == 0) {
        float t = (lane < nw) ? red[lane] : 0.0f;
        t = warpReduceSum(t);
        if (lane == 0) red[0] = t;
    }
    __syncthreads();
    return red[0];
}

__device__ __forceinline__ float siluf(float x) {
    return x * (1.0f / (1.0f + __expf(-x)));
}

// ---------------- CDNA5 async global->LDS copy (16 B per lane) --------------
__device__ __forceinline__ void async_copy_b128(void* lds_ptr, const float* gptr) {
    const unsigned int       lo = (unsigned int)(unsigned long long)lds_ptr; // LDS offset
    const unsigned long long ga = (unsigned long long)gptr;
    asm volatile("global_load_async_to_lds_b128 %0, %1, off"
                 :: "v"(lo), "v"(ga) : "memory");
}

__device__ __forceinline__ void wait_async0() {
    asm volatile("s_wait_asynccnt 0" ::: "memory");
}

// ---------------- NT GEMM: double-buffered LDS panels + f32 WMMA ------------
// C[m,n] = act( sum_k A[m,k]*B[n,k] + bias[n] ) + res[m,n]
// Block: 256 threads = 8 waves; strip = 16 rows x 128 cols; K chunked by 64.
// Requires: M % 16 == 0, K % 4 == 0, all row strides % 4 == 0 (true here).
__global__ __launch_bounds__(256)
void wmma_gemm_nt(const float* __restrict__ A, const float* __restrict__ B,
                  const float* __restrict__ bias, const float* __restrict__ res,
                  float* __restrict__ C,
                  int M, int N, int K, int lda, int ldb, int ldc, int op)
{
    __shared__ __align__(16) float As[2][16  * LDSS];   // 2 x  ~4.3 KB
    __shared__ __align__(16) float Bs[2][BNq * LDSS];   // 2 x ~34.8 KB

    const int tid  = threadIdx.x;
    const int lane = tid & 31;
    const int wid  = tid >> 5;          // 0..7
    const int half = lane >> 4;         // 0: k,k+1   1: k+2,k+3
    const int lm   = lane & 15;

    const int m0     = blockIdx.y * 16;
    const int nbase  = blockIdx.x * BNq;
    const int n0     = nbase + wid * 16;
    const int col    = n0 + lm;
    const bool bval  = (col < N);

    // copy-phase mapping: each thread owns one 4-float slot of As and 8 of Bs
    const int arow_l = tid >> 4;                 // 0..15
    const int akk    = (tid & 15) * 4;           // 0..60

    // Stage K-chunk k0 into buffer 'buf'. Out-of-range (K tail / N edge) slots
    // are zero-filled with DS stores; addresses are disjoint from the async
    // writes, so the DScnt-vs-ASYNCcnt ordering hazard cannot occur.
    auto stage = [&](int buf, int k0) {
        const int kc = (K - k0 < KCq) ? (K - k0) : KCq;   // multiple of 4
        const float4 z = {0.f, 0.f, 0.f, 0.f};
        if (akk < kc)
            async_copy_b128(&As[buf][arow_l * LDSS + akk],
                            A + (size_t)(m0 + arow_l) * lda + k0 + akk);
        else
            *(float4*)&As[buf][arow_l * LDSS + akk] = z;
#pragma unroll
        for (int it = 0; it < 8; ++it) {
            const int linear = it * 1024 + tid * 4;
            const int brow = linear >> 6;
            const int bkk  = linear & 63;
            const int gcol = nbase + brow;
            if (bkk < kc && gcol < N)
                async_copy_b128(&Bs[buf][brow * LDSS + bkk],
                                B + (size_t)gcol * ldb + k0 + bkk);
            else
                *(float4*)&Bs[buf][brow * LDSS + bkk] = z;
        }
    };

    v8f acc = {0.f, 0.f, 0.f, 0.f, 0.f, 0.f, 0.f, 0.f};
    const int nch = (K + KCq - 1) / KCq;

    stage(0, 0);
    for (int c = 0; c < nch; ++c) {
        const int cur = c & 1;
        wait_async0();        // panel 'cur' (staged last iteration) is in LDS
        __syncthreads();      // visible to all waves; prev reads of it done
        if (c + 1 < nch) stage(1 - cur, (c + 1) * KCq);   // overlap with WMMAs

#pragma unroll
        for (int kk = 0; kk < KCq; kk += 4) {
            v2f a = *(const v2f*)&As[cur][lm * LDSS + kk + 2 * half];
            v2f b = *(const v2f*)&Bs[cur][(wid * 16 + lm) * LDSS + kk + 2 * half];
            acc = __builtin_amdgcn_wmma_f32_16x16x4_f32(
                false, a, false, b, (short)0, acc, false, false);
        }
        __syncthreads();      // all reads of 'cur' done before restage at c+2
    }

    if (bval) {
        const float bvv = bias ? bias[col] : 0.0f;
#pragma unroll
        for (int r = 0; r < 8; ++r) {
            const int row = m0 + r + 8 * half;
            float v = acc[r] + bvv;
            if (op == OP_SOFTPLUS) v = (v > 20.0f) ? v : log1pf(__expf(v));
            if (res) v += res[(size_t)row * ldc + col];
            C[(size_t)row * ldc + col] = v;
        }
    }
}

// ---------------- LayerNorm -------------------------------------------------
__global__ void ln_kernel(const float* __restrict__ x, const float* __restrict__ w,
                          const float* __restrict__ b, float* __restrict__ o, int D)
{
    __shared__ float red[32];
    const int row = blockIdx.x;
    const float* xr = x + (size_t)row * D;
    float* orow = o + (size_t)row * D;

    float s = 0.0f;
    for (int i = threadIdx.x; i < D; i += blockDim.x) s += xr[i];
    const float mean = blockReduceSum(s, red) / (float)D;

    float v = 0.0f;
    for (int i = threadIdx.x; i < D; i += blockDim.x) {
        const float d = xr[i] - mean;
        v += d * d;
    }
    const float rstd = rsqrtf(blockReduceSum(v, red) / (float)D + 1e-5f);

    for (int i = threadIdx.x; i < D; i += blockDim.x)
        orow[i] = (xr[i] - mean) * rstd * w[i] + b[i];
}

// ---------------- depthwise causal conv (CONV=4) + SiLU ---------------------
__global__ void conv_silu_kernel(const float* __restrict__ u, const float* __restrict__ cw,
                                 const float* __restrict__ cb, float* __restrict__ xconv,
                                 int ldu)
{
    const int idx = blockIdx.x * blockDim.x + threadIdx.x;
    if (idx >= Lq * DIq) return;
    const int d = idx % DIq;
    const int l = idx / DIq;
    float acc = cb[d];
#pragma unroll
    for (int j = 0; j < 4; ++j) {
        const int lp = l - 3 + j;
        if (lp >= 0) acc += cw[j * DIq + d] * u[(size_t)lp * ldu + d];
    }
    xconv[idx] = siluf(acc);
}

// ---------------- selective scan: one wave per channel d --------------------
__global__ __launch_bounds__(32)
void mamba_scan_kernel(const float* __restrict__ dt,     // (L, DI)
                       const float* __restrict__ xconv,  // (L, DI)
                       const float* __restrict__ dbl,    // (L, DBL): [dt|B|C]
                       const float* __restrict__ A_log,  // (DI, DS)
                       const float* __restrict__ Dp,     // (DI)
                       const float* __restrict__ u,      // z at u[:, DI:], ldu = 2*DI
                       float* __restrict__ y)            // (L, DI)
{
    const int d    = blockIdx.x;
    const int lane = threadIdx.x;

    float Aneg[4], h[4] = {0.f, 0.f, 0.f, 0.f};
#pragma unroll
    for (int j = 0; j < 4; ++j)
        Aneg[j] = -__expf(A_log[(size_t)d * DSq + lane + 32 * j]);
    const float dpv = Dp[d];

    for (int l = 0; l < Lq; ++l) {
        const float dtv = dt[(size_t)l * DIq + d];       // uniform across wave
        const float xv  = xconv[(size_t)l * DIq + d];
        const float* row = dbl + (size_t)l * DBLq;
        float acc = 0.0f;
#pragma unroll
        for (int j = 0; j < 4; ++j) {
            const int s  = lane + 32 * j;
            const float dA = __expf(dtv * Aneg[j]);
            const float bx = dtv * row[DTRq + s] * xv;   // dt * B[s] * x
            h[j] = dA * h[j] + bx;
            acc += h[j] * row[DTRq + DSq + s];           // h * C[s]
        }
        acc = warpReduceSum(acc);
        if (lane == 0) {
            const float zv = u[(size_t)l * (2 * DIq) + DIq + d];
            y[(size_t)l * DIq + d] = (acc + dpv * xv) * siluf(zv);
        }
    }
}

// ---------------- sliding-window attention (WIN=8, NH=4, DH=80) -------------
__global__ __launch_bounds__(32)
void local_attn_kernel(const float* __restrict__ qkv, float* __restrict__ o)
{
    const int l    = blockIdx.x;
    const int hd   = blockIdx.y;
    const int lane = threadIdx.x;
    const float scale = 0.1118033988749895f;  // 1/sqrt(80)

    const float* q = qkv + (size_t)l * (3 * DMq) + hd * DHq;

    float sc[WINq];
#pragma unroll
    for (int w = 0; w < WINq; ++w) {
        const int lp = l - (WINq - 1) + w;
        float s = -1.0e9f;
        if (lp >= 0) {
            const float* k = qkv + (size_t)lp * (3 * DMq) + DMq + hd * DHq;
            float part = 0.0f;
            for (int dd = lane; dd < DHq; dd += 32) part += q[dd] * k[dd];
            s = warpReduceSum(part) * scale;
        }
        sc[w] = s;
    }
    float mx = sc[0];
#pragma unroll
    for (int w = 1; w < WINq; ++w) mx = fmaxf(mx, sc[w]);
    float se = 0.0f;
#pragma unroll
    for (int w = 0; w < WINq; ++w) { sc[w] = __expf(sc[w] - mx); se += sc[w]; }
    const float inv = 1.0f / se;

    for (int dd = lane; dd < DHq; dd += 32) {
        float acc = 0.0f;
#pragma unroll
        for (int w = 0; w < WINq; ++w) {
            const int lp = l - (WINq - 1) + w;
            if (lp >= 0)
                acc += sc[w] * qkv[(size_t)lp * (3 * DMq) + 2 * DMq + hd * DHq + dd];
        }
        o[(size_t)l * DMq + hd * DHq + dd] = acc * inv;
    }
}

// ---------------- row-wise L2 normalize -------------------------------------
__global__ void l2norm_kernel(float* __restrict__ o, int D)
{
    __shared__ float red[32];
    const int row = blockIdx.x;
    float* r = o + (size_t)row * D;
    float s = 0.0f;
    for (int i = threadIdx.x; i < D; i += blockDim.x) { const float v = r[i]; s += v * v; }
    s = blockReduceSum(s, red);
    const float inv = 1.0f / fmaxf(sqrtf(s), 1e-12f);
    for (int i = threadIdx.x; i < D; i += blockDim.x) r[i] *= inv;
}

// ---------------------------------------------------------------------------
static inline void gemm(hipStream_t st, const float* A, const float* B,
                        const float* bias, const float* res, float* C,
                        int M, int N, int K, int lda, int ldb, int ldc, int op)
{
    dim3 grid((N + BNq - 1) / BNq, M / 16);
    wmma_gemm_nt<<<grid, 256, 0, st>>>(A, B, bias, res, C, M, N, K, lda, ldb, ldc, op);
}

extern "C" void kernel_launch(void* const* d_in, const int* in_sizes, int n_in,
                              void* d_out, int out_size, void* d_ws, size_t ws_size,
                              hipStream_t stream)
{
    (void)in_sizes; (void)n_in; (void)out_size; (void)ws_size;

    const float* x      = (const float*)d_in[0];   // (512,768)
    const float* W_in   = (const float*)d_in[1];   // (320,768)
    const float* b_in   = (const float*)d_in[2];   // (320)
    const float* W_head = (const float*)d_in[3];   // (768,320)
    const float* ln_w   = (const float*)d_in[4];   // (9,320)
    const float* ln_b   = (const float*)d_in[5];
    const float* Win    = (const float*)d_in[6];   // (9,1280,320)
    const float* conv_w = (const float*)d_in[7];   // (9,4,640)
    const float* conv_b = (const float*)d_in[8];   // (9,640)
    const float* Wx     = (const float*)d_in[9];   // (9,276,640)
    const float* Wdt    = (const float*)d_in[10];  // (9,640,20)
    const float* bdt    = (const float*)d_in[11];  // (9,640)
    const float* A_log  = (const float*)d_in[12];  // (9,640,128)
    const float* Dp     = (const float*)d_in[13];  // (9,640)
    const float* Wout   = (const float*)d_in[14];  // (9,320,640)
    const float* aln_w  = (const float*)d_in[15];  // (3,320)
    const float* aln_b  = (const float*)d_in[16];
    const float* Wqkv   = (const float*)d_in[17];  // (3,960,320)
    const float* bqkv   = (const float*)d_in[18];  // (3,960)
    const float* Wo     = (const float*)d_in[19];  // (3,320,320)
    const float* bo     = (const float*)d_in[20];  // (3,320)

    // workspace layout (fp32), ~8.4 MB total
    float* h     = (float*)d_ws;                 // 512*320
    float* xln   = h     + (size_t)Lq * DMq;     // 512*320
    float* u     = xln   + (size_t)Lq * DMq;     // 512*1280  (also qkv 512*960)
    float* xconv = u     + (size_t)Lq * 2 * DIq; // 512*640   (also attn 'o' 512*320)
    float* dblb  = xconv + (size_t)Lq * DIq;     // 512*276
    float* dtb   = dblb  + (size_t)Lq * DBLq;    // 512*640
    float* yb    = dtb   + (size_t)Lq * DIq;     // 512*640
    float* out   = (float*)d_out;                // (512,768)

    // stem: h = x @ W_in.T + b_in
    gemm(stream, x, W_in, b_in, nullptr, h, Lq, DMq, 768, 768, 768, DMq, OP_NONE);

    int mi = 0, ai = 0;
    for (int i = 0; i < 12; ++i) {
        if ((i + 1) % 4 == 0) {
            // ---- attention layer ----
            ln_kernel<<<Lq, 320, 0, stream>>>(h, aln_w + ai * DMq, aln_b + ai * DMq, xln, DMq);
            gemm(stream, xln, Wqkv + (size_t)ai * 3 * DMq * DMq, bqkv + ai * 3 * DMq,
                 nullptr, u, Lq, 3 * DMq, DMq, DMq, DMq, 3 * DMq, OP_NONE);
            local_attn_kernel<<<dim3(Lq, NHq), 32, 0, stream>>>(u, xconv);
            gemm(stream, xconv, Wo + (size_t)ai * DMq * DMq, bo + ai * DMq,
                 h, h, Lq, DMq, DMq, DMq, DMq, DMq, OP_NONE);
            ai++;
        } else {
            // ---- mamba layer ----
            ln_kernel<<<Lq, 320, 0, stream>>>(h, ln_w + mi * DMq, ln_b + mi * DMq, xln, DMq);
            // u = [xc | z] = xln @ Win.T
            gemm(stream, xln, Win + (size_t)mi * 2 * DIq * DMq, nullptr, nullptr,
                 u, Lq, 2 * DIq, DMq, DMq, DMq, 2 * DIq, OP_NONE);
            conv_silu_kernel<<<(Lq * DIq + 255) / 256, 256, 0, stream>>>(
                u, conv_w + (size_t)mi * 4 * DIq, conv_b + mi * DIq, xconv, 2 * DIq);
            // dbl = xconv @ Wx.T
            gemm(stream, xconv, Wx + (size_t)mi * DBLq * DIq, nullptr, nullptr,
                 dblb, Lq, DBLq, DIq, DIq, DIq, DBLq, OP_NONE);
            // dt = softplus(dbl[:, :20] @ Wdt.T + bdt)
            gemm(stream, dblb, Wdt + (size_t)mi * DIq * DTRq, bdt + mi * DIq, nullptr,
                 dtb, Lq, DIq, DTRq, DBLq, DTRq, DIq, OP_SOFTPLUS);
            // sequential scan + D-skip + SiLU(z) gate
            mamba_scan_kernel<<<DIq, 32, 0, stream>>>(
                dtb, xconv, dblb, A_log + (size_t)mi * DIq * DSq, Dp + mi * DIq, u, yb);
            // h = h + y @ Wout.T
            gemm(stream, yb, Wout + (size_t)mi * DMq * DIq, nullptr, h,
                 h, Lq, DMq, DIq, DIq, DIq, DMq, OP_NONE);
            mi++;
        }
    }

    // head: out = h @ W_head.T, then L2 normalize rows
    gemm(stream, h, W_head, nullptr, nullptr, out, Lq, 768, DMq, DMq, DMq, 768, OP_NONE);
    l2norm_kernel<<<Lq, 256, 0, stream>>>(out, 768);
}